// ctc_hinge_loss_5660766896275
// MI455X (gfx1250) — compile-verified
//
#include <hip/hip_runtime.h>
#include <stdint.h>

// Problem constants (static per reference setup_inputs)
#define T_DIM 1500
#define B_DIM 96
#define V_DIM 128
#define L_GT  256
#define L_CAP 512
#define NEGF  (-1e30f)
#define K_LOG2E 1.4426950408889634f
#define K_LN2   0.6931471805599453f

// Workspace layout (bytes), total ~1.35 MB:
#define WS_LSE    0
#define WS_PRED   576000
#define WS_DEC    1152000
#define WS_DECLEN 1348608
#define WS_COSTS  1349120

// ---------------------------------------------------------------------------
// Kernel A: per-(t,b) row logsumexp + argmax. One wave32 per 128-wide row.
// Single HBM pass over acts (73.7 MB): bandwidth-bound, a few us on MI455X.
// ---------------------------------------------------------------------------
__global__ __launch_bounds__(256) void k_rowstats(const float* __restrict__ acts,
                                                  float* __restrict__ lse,
                                                  int* __restrict__ pred) {
  int wave = (blockIdx.x * blockDim.x + threadIdx.x) >> 5;
  int lane = threadIdx.x & 31;
  if (wave >= T_DIM * B_DIM) return;
  const float4 x4 = reinterpret_cast<const float4*>(acts + (size_t)wave * V_DIM)[lane];

  // argmax with first-occurrence tie-break (matches jnp.argmax)
  float bv = x4.x; int bi = lane * 4;
  if (x4.y > bv) { bv = x4.y; bi = lane * 4 + 1; }
  if (x4.z > bv) { bv = x4.z; bi = lane * 4 + 2; }
  if (x4.w > bv) { bv = x4.w; bi = lane * 4 + 3; }
  for (int off = 16; off > 0; off >>= 1) {
    float ov = __shfl_xor(bv, off, 32);
    int   oi = __shfl_xor(bi, off, 32);
    if (ov > bv || (ov == bv && oi < bi)) { bv = ov; bi = oi; }
  }
  // stable logsumexp (natural log)
  float s = __expf(x4.x - bv) + __expf(x4.y - bv) + __expf(x4.z - bv) + __expf(x4.w - bv);
  for (int off = 16; off > 0; off >>= 1) s += __shfl_xor(s, off, 32);
  if (lane == 0) {
    lse[wave]  = bv + __logf(s);
    pred[wave] = bi;
  }
}

// ---------------------------------------------------------------------------
// Kernel B: greedy CTC collapse per batch element. One wave per b; ballot +
// popc wave-prefix-scan, 47 iterations of 32 timesteps.
// ---------------------------------------------------------------------------
__global__ __launch_bounds__(32) void k_decode(const int* __restrict__ pred,
                                               const int* __restrict__ act_lens,
                                               int* __restrict__ dec,
                                               int* __restrict__ dec_lens) {
  int b = blockIdx.x;
  int lane = threadIdx.x;
  for (int i = lane; i < L_CAP; i += 32) dec[b * L_CAP + i] = 0;  // zero pad (blank)
  int Tact = act_lens[b];
  int count = 0;
  for (int base = 0; base < Tact; base += 32) {
    int t = base + lane;
    int pv = 0, prev = -1;
    if (t < Tact) {
      pv = pred[(size_t)t * B_DIM + b];
      prev = (t == 0) ? -1 : pred[(size_t)(t - 1) * B_DIM + b];
    }
    bool nt = (pv != 0) && (pv != prev);           // pv==0 covers t>=Tact lanes
    unsigned mask = (unsigned)__ballot(nt);
    int pos = count + __popc(mask & ((1u << lane) - 1u));
    if (nt && pos < L_CAP) dec[b * L_CAP + pos] = pv;
    count += __popc(mask);
  }
  if (lane == 0) {
    int dl = count; if (dl > L_CAP) dl = L_CAP; if (dl < 1) dl = 1;
    dec_lens[b] = dl;
  }
}

// ---------------------------------------------------------------------------
// Kernel C: CTC alpha recursion in log2 domain, one block per batch element,
// templated on label capacity L so S and trip counts are compile-time.
//  - every wave issues its OWN async copies (rows + lse chunks) and waits its
//    own ASYNCcnt => only ONE __syncthreads per timestep (alpha flip).
//  - acts row (512B) double-buffered via global_load_async_to_lds_b128
//  - lse values gathered 32-ahead via global_load_async_to_lds_b32
// ---------------------------------------------------------------------------
#define BD 256

template <int L>
__global__ __launch_bounds__(BD) void k_ctc(const float* __restrict__ acts,
                                            const int* __restrict__ lab_all,   // [B][L]
                                            const int* __restrict__ act_lens,
                                            const int* __restrict__ lab_lens,  // [B]
                                            const float* __restrict__ lse,
                                            float* __restrict__ cost) {        // [B]
  constexpr int S = 2 * L + 1;
  constexpr int KMAX = (S + BD - 1) / BD;
  __shared__ float alpha[2][S + 1];
  __shared__ float rowbuf[2][V_DIM];
  __shared__ float lsebuf[2][32];

  const int b    = blockIdx.x;
  const int tid  = threadIdx.x;
  const int lane = tid & 31;
  const int* lab = lab_all + b * L;
  const int Llen = lab_lens[b];
  const int Tact = act_lens[b];

  // Per-thread ext label / skip flag, fully unrolled into registers.
  int  exts[KMAX];
  bool skips[KMAX];
#pragma unroll
  for (int k = 0; k < KMAX; ++k) {
    int s = tid + k * BD;
    int e = 0; bool sk = false;
    if (s < S && (s & 1)) {
      int i = s >> 1;
      e = lab[i];
      sk = (e != 0) && ((i == 0) || (e != lab[i - 1])) && (s >= 2);
    }
    exts[k] = e; skips[k] = sk;
  }

#pragma unroll
  for (int k = 0; k < KMAX; ++k) {
    int s = tid + k * BD;
    if (s < S) alpha[0][s] = NEGF;
  }
  if (tid == 0) {
    float lse0 = lse[b];
    alpha[0][0] = (acts[(size_t)b * V_DIM]          - lse0) * K_LOG2E;
    alpha[0][1] = (acts[(size_t)b * V_DIM + lab[0]] - lse0) * K_LOG2E;
  }

  // LDS byte bases (generic-pointer low 32 bits == LDS offset)
  const unsigned rowBase = (unsigned)(unsigned long long)(const void*)&rowbuf[0][0];
  const unsigned lseBase = (unsigned)(unsigned long long)(const void*)&lsebuf[0][0];

  // Prologue: every wave issues chunk0, chunk1, row1 (in-order completion).
  {
    unsigned d0 = lseBase + lane * 4u;
    unsigned long long g0 = (unsigned long long)(const void*)(lse + (size_t)lane * B_DIM + b);
    asm volatile("global_load_async_to_lds_b32 %0, %1, off" :: "v"(d0), "v"(g0) : "memory");
    unsigned d1 = lseBase + 128u + lane * 4u;
    unsigned long long g1 = (unsigned long long)(const void*)(lse + (size_t)(32 + lane) * B_DIM + b);
    asm volatile("global_load_async_to_lds_b32 %0, %1, off" :: "v"(d1), "v"(g1) : "memory");
    unsigned dr = rowBase + (unsigned)(V_DIM * 4) + lane * 16u;
    unsigned long long gr = (unsigned long long)(const void*)
        (acts + ((size_t)1 * B_DIM + b) * V_DIM + lane * 4);
    asm volatile("global_load_async_to_lds_b128 %0, %1, off" :: "v"(dr), "v"(gr) : "memory");
  }
  __syncthreads();

  int cur = 0;
  for (int t = 1; t < Tact; ++t) {
    const bool row_iss   = (t + 1 < Tact);
    const bool chunk_iss = ((t & 31) == 0) && (t + 32 < Tact);
    if (chunk_iss) {
      int c1 = (t >> 5) + 1;                       // next lse chunk
      int tl = c1 * 32 + lane; if (tl > T_DIM - 1) tl = T_DIM - 1;
      unsigned d = lseBase + (unsigned)((c1 & 1) * 128) + lane * 4u;
      unsigned long long g = (unsigned long long)(const void*)(lse + (size_t)tl * B_DIM + b);
      asm volatile("global_load_async_to_lds_b32 %0, %1, off" :: "v"(d), "v"(g) : "memory");
    }
    if (row_iss) {
      unsigned d = rowBase + (unsigned)(((t + 1) & 1) * (V_DIM * 4)) + lane * 16u;
      unsigned long long g = (unsigned long long)(const void*)
          (acts + ((size_t)(t + 1) * B_DIM + b) * V_DIM + lane * 4);
      asm volatile("global_load_async_to_lds_b128 %0, %1, off" :: "v"(d), "v"(g) : "memory");
    }
    // Need row t complete; ops issued after it this step: chunk? + row?
    if (row_iss && chunk_iss)      asm volatile("s_wait_asynccnt 0x2" ::: "memory");
    else if (row_iss || chunk_iss) asm volatile("s_wait_asynccnt 0x1" ::: "memory");
    else                           asm volatile("s_wait_asynccnt 0x0" ::: "memory");

    const float lse2 = lsebuf[(t >> 5) & 1][t & 31] * K_LOG2E;
    const float* __restrict__ aOld = alpha[cur];
    float* __restrict__ aNew = alpha[cur ^ 1];
    const float* __restrict__ rw = rowbuf[t & 1];
#pragma unroll
    for (int k = 0; k < KMAX; ++k) {
      int s = tid + k * BD;
      if (s < S) {
        float a0 = aOld[s];
        float a1 = (s > 0)  ? aOld[s - 1] : NEGF;
        float a2 = skips[k] ? aOld[s - 2] : NEGF;
        float m  = fmaxf(a0, fmaxf(a1, a2));
        float sum = __builtin_amdgcn_exp2f(a0 - m) + __builtin_amdgcn_exp2f(a1 - m)
                  + __builtin_amdgcn_exp2f(a2 - m);
        aNew[s] = m + __builtin_amdgcn_logf(sum) + (rw[exts[k]] * K_LOG2E - lse2);
      }
    }
    cur ^= 1;
    __syncthreads();   // single barrier per step: alpha flip + buffer recycling
  }

  if (tid == 0) {
    float aL  = alpha[cur][2 * Llen];
    float aL1 = alpha[cur][2 * Llen - 1];
    float m  = fmaxf(aL, aL1);
    float e2 = m + __builtin_amdgcn_logf(__builtin_amdgcn_exp2f(aL - m)
                                       + __builtin_amdgcn_exp2f(aL1 - m));
    cost[b] = -e2 * K_LN2;   // back to natural log
  }
}

// ---------------------------------------------------------------------------
// Kernel D: final combine out = cost_gt - cost_hyp + 1
// ---------------------------------------------------------------------------
__global__ void k_combine(const float* __restrict__ costs, float* __restrict__ out) {
  int i = threadIdx.x + blockIdx.x * blockDim.x;
  if (i < B_DIM) out[i] = costs[i] - costs[B_DIM + i] + 1.0f;
}

extern "C" void kernel_launch(void* const* d_in, const int* in_sizes, int n_in,
                              void* d_out, int out_size, void* d_ws, size_t ws_size,
                              hipStream_t stream) {
  const float* acts      = (const float*)d_in[0];
  const int*  labels     = (const int*)d_in[1];
  const int*  act_lens   = (const int*)d_in[2];
  const int*  label_lens = (const int*)d_in[3];

  char* ws = (char*)d_ws;
  float* lse      = (float*)(ws + WS_LSE);
  int*   pred     = (int*)(ws + WS_PRED);
  int*   dec      = (int*)(ws + WS_DEC);
  int*   dec_lens = (int*)(ws + WS_DECLEN);
  float* costs    = (float*)(ws + WS_COSTS);
  float* out      = (float*)d_out;

  // A: 144000 rows, 8 waves/block -> 18000 blocks
  k_rowstats<<<(T_DIM * B_DIM) / 8, 256, 0, stream>>>(acts, lse, pred);
  // B: one wave per batch element
  k_decode<<<B_DIM, 32, 0, stream>>>(pred, act_lens, dec, dec_lens);
  // C: ground-truth CTC (S=513) and hypothesis CTC (S=1025), one block per b
  k_ctc<L_GT ><<<B_DIM, BD, 0, stream>>>(acts, labels, act_lens, label_lens, lse, costs);
  k_ctc<L_CAP><<<B_DIM, BD, 0, stream>>>(acts, dec,    act_lens, dec_lens,   lse, costs + B_DIM);
  // D: combine
  k_combine<<<1, 128, 0, stream>>>(costs, out);
}